// Detection_Loss_37391985279244
// MI455X (gfx1250) — compile-verified
//
#include <hip/hip_runtime.h>

// Problem constants (match reference)
#define B_   16
#define N_   8192
#define M_   64
#define ROW_ 85          // 4 box + 1 obj + 80 cls
#define SCALE_     640.0f
#define IOU_MATCH_ 0.5f
#define NMS_THR_   0.5f
#define EPS_       1e-7f
#define PROB_T_    0.0f

typedef __attribute__((ext_vector_type(2))) float v2f;
typedef __attribute__((ext_vector_type(8))) float v8f;

// ---------------------------------------------------------------------------
// CDNA5 async global->LDS copy (ASYNCcnt path). 16 bytes per active lane.
// ldsOff is the byte offset within the workgroup's LDS allocation: generic
// pointers to LDS are {aperture_hi, lds_offset}, so truncation recovers it.
// ---------------------------------------------------------------------------
__device__ __forceinline__ unsigned lds_offset(const void* p) {
    return (unsigned)(uintptr_t)p;
}
__device__ __forceinline__ void async_copy_b128(unsigned ldsOff, const void* gptr) {
    asm volatile("global_load_async_to_lds_b128 %0, %1, off"
                 :: "v"(ldsOff), "v"((unsigned long long)(uintptr_t)gptr)
                 : "memory");
}
__device__ __forceinline__ void wait_asynccnt0() {
    asm volatile("s_wait_asynccnt 0" ::: "memory");
}

// ---------------------------------------------------------------------------
// Kernel 1: per-prediction prep.
//   pred box (xyxy), key = valid ? (cls0*obj) : -1
//   valid = (obj >= PROB_T) && any_m IoU(gt_m, pred) >= 0.5
// Grid: 512 blocks x 256 threads; 32 blocks per image. GT boxes staged in LDS.
// ---------------------------------------------------------------------------
__global__ void det_prep_kernel(const float* __restrict__ out,
                                const float* __restrict__ lab,
                                float4* __restrict__ ws_boxes,
                                float*  __restrict__ ws_key) {
    __shared__ float gx1[M_], gy1[M_], gx2[M_], gy2[M_], garea[M_];
    const int img = blockIdx.x >> 5;                     // 32 blocks / image
    const int j   = ((blockIdx.x & 31) << 8) + threadIdx.x;

    if (threadIdx.x < M_) {
        const float* lp = lab + (img * M_ + threadIdx.x) * 5 + 1; // skip cls
        float cx = lp[0], cy = lp[1], w = lp[2], h = lp[3];
        float x1 = fminf(fmaxf(cx - 0.5f * w, 0.f), 1.f) * SCALE_;
        float y1 = fminf(fmaxf(cy - 0.5f * h, 0.f), 1.f) * SCALE_;
        float x2 = fminf(fmaxf(cx + 0.5f * w, 0.f), 1.f) * SCALE_;
        float y2 = fminf(fmaxf(cy + 0.5f * h, 0.f), 1.f) * SCALE_;
        gx1[threadIdx.x] = x1; gy1[threadIdx.x] = y1;
        gx2[threadIdx.x] = x2; gy2[threadIdx.x] = y2;
        garea[threadIdx.x] = (x2 - x1) * (y2 - y1);
    }
    __syncthreads();

    const float* op = out + ((size_t)img * N_ + j) * ROW_;
    float px = op[0], py = op[1], pw = op[2], ph = op[3];
    float obj = op[4], cls0 = op[5];

    float x1 = px - 0.5f * pw, y1 = py - 0.5f * ph;
    float x2 = px + 0.5f * pw, y2 = py + 0.5f * ph;
    float parea = (x2 - x1) * (y2 - y1);

    bool match = false;
    for (int m = 0; m < M_; ++m) {
        float ix1 = fmaxf(gx1[m], x1), iy1 = fmaxf(gy1[m], y1);
        float ix2 = fminf(gx2[m], x2), iy2 = fminf(gy2[m], y2);
        float iw = fmaxf(ix2 - ix1, 0.f), ih = fmaxf(iy2 - iy1, 0.f);
        float inter = iw * ih;
        float iou = inter / (garea[m] + parea - inter);   // reference box_iou: no eps
        match = match || (iou >= IOU_MATCH_);
    }
    bool valid = (obj >= PROB_T_) && match;
    float logit = cls0 * obj;                             // >= 0 always

    int gi = img * N_ + j;
    ws_boxes[gi] = make_float4(x1, y1, x2, y2);
    ws_key[gi]   = valid ? logit : -1.0f;                 // invalid sinks to tail
}

// ---------------------------------------------------------------------------
// Kernel 2: per-image bitonic sort (descending by key) of 8192 (key, idx)
// pairs entirely in LDS (64 KB); keys staged via async global->LDS copies,
// then gather sorted boxes/keys to global.
// ---------------------------------------------------------------------------
__global__ void __launch_bounds__(1024)
det_sort_kernel(const float*  __restrict__ ws_key,
                const float4* __restrict__ ws_boxes,
                float*  __restrict__ ws_skey,
                float4* __restrict__ ws_sboxes) {
    __shared__ float sk[N_];
    __shared__ int   si[N_];
    const int img = blockIdx.x;
    const int tid = threadIdx.x;

    // Stage keys with the CDNA5 async-to-LDS engine: 2 x b128 per thread.
    {
        unsigned base = lds_offset(&sk[0]);
        const char* g = (const char*)(ws_key + img * N_);
        async_copy_b128(base + tid * 16,         g + tid * 16);
        async_copy_b128(base + tid * 16 + 16384, g + tid * 16 + 16384);
    }
    for (int e = 0; e < 8; ++e) si[e * 1024 + tid] = e * 1024 + tid;
    wait_asynccnt0();
    __syncthreads();

    for (int k = 2; k <= N_; k <<= 1) {
        for (int s = k >> 1; s > 0; s >>= 1) {
            __syncthreads();
            for (int e = 0; e < 8; ++e) {
                int i = e * 1024 + tid;
                int p = i ^ s;
                if (p > i) {
                    float ki = sk[i], kp = sk[p];
                    bool desc = ((i & k) == 0);           // descending blocks
                    bool sw = desc ? (ki < kp) : (ki > kp);
                    if (sw) {
                        sk[i] = kp; sk[p] = ki;
                        int t = si[i]; si[i] = si[p]; si[p] = t;
                    }
                }
            }
        }
    }
    __syncthreads();

    for (int e = 0; e < 8; ++e) {
        int p = e * 1024 + tid;
        int src = si[p];
        ws_skey[img * N_ + p]   = sk[p];
        ws_sboxes[img * N_ + p] = ws_boxes[img * N_ + src];
    }
}

// ---------------------------------------------------------------------------
// Kernel 3: per-image greedy NMS + reduction. One 1024-thread WG per image.
// Sorted keys live in LDS (async-staged, 32 KB); suppression is a 1 KB LDS
// bitmask. The serial chain only pays a block-wide round per *active* index:
// thread 0 skips suppressed indices word-at-a-time via ~supp + ffs, so the
// number of barrier rounds ~= number of NMS survivors (O(100)), not 8192.
// Broadcast box reads are global but L2-resident. Final sums use
// v_wmma_f32_16x16x4_f32 with an all-ones B matrix (exact fp32 reduction;
// slot assignment is an arbitrary bijection since summation commutes).
// ---------------------------------------------------------------------------
__global__ void __launch_bounds__(1024)
det_nms_kernel(const float*  __restrict__ ws_skey,
               const float4* __restrict__ ws_sboxes,
               float* __restrict__ loss_out) {
    __shared__ float    lkey[N_];          // 32 KB sorted keys
    __shared__ unsigned supp[N_ / 32];     // suppression bitmask
    __shared__ float    bc[5];             // broadcast box + area
    __shared__ int      s_cur;             // active index, or -1 when done
    __shared__ float    psum[1024];
    __shared__ float    pcnt[1024];

    const int img = blockIdx.x;
    const int tid = threadIdx.x;
    const float4* sbox = ws_sboxes + img * N_;

    // Async-stage sorted keys into LDS: 2 x b128 per thread.
    {
        unsigned base = lds_offset(&lkey[0]);
        const char* g = (const char*)(ws_skey + img * N_);
        async_copy_b128(base + tid * 16,         g + tid * 16);
        async_copy_b128(base + tid * 16 + 16384, g + tid * 16 + 16384);
    }

    // Own 8 boxes in VGPRs (coalesced one-time global read, L2-resident).
    float4 myb[8]; float mya[8];
    for (int e = 0; e < 8; ++e) {
        int j = e * 1024 + tid;
        myb[e] = sbox[j];
        mya[e] = (myb[e].z - myb[e].x + 1.f) * (myb[e].w - myb[e].y + 1.f);
    }
    if (tid < N_ / 32) supp[tid] = 0u;
    wait_asynccnt0();
    __syncthreads();

    float myk[8];
    for (int e = 0; e < 8; ++e) myk[e] = lkey[e * 1024 + tid];

    int i = 0;
    while (true) {
        if (tid == 0) {
            // Skip suppressed indices word-at-a-time.
            int ii = i;
            while (ii < N_) {
                unsigned w = ~supp[ii >> 5] & (0xFFFFFFFFu << (ii & 31));
                if (w) { ii = (ii & ~31) + (__ffs(w) - 1); break; }
                ii = (ii & ~31) + 32;
            }
            if (ii >= N_ || lkey[ii] < 0.f) {   // sorted: rest are invalid
                s_cur = -1;
            } else {
                s_cur = ii;
                float4 bi = sbox[ii];
                bc[0] = bi.x; bc[1] = bi.y; bc[2] = bi.z; bc[3] = bi.w;
                bc[4] = (bi.z - bi.x + 1.f) * (bi.w - bi.y + 1.f);
            }
        }
        __syncthreads();
        int ci = s_cur;
        if (ci < 0) break;

        float xi1 = bc[0], yi1 = bc[1], xi2 = bc[2], yi2 = bc[3], ai = bc[4];
        for (int e = 0; e < 8; ++e) {
            int j = e * 1024 + tid;
            if (j > ci) {
                float xx1 = fmaxf(xi1, myb[e].x);
                float yy1 = fmaxf(yi1, myb[e].y);
                float xx2 = fminf(xi2, myb[e].z);
                float yy2 = fminf(yi2, myb[e].w);
                float w = fmaxf(xx2 - xx1 + 1.f, 0.f);
                float h = fmaxf(yy2 - yy1 + 1.f, 0.f);
                float inter = w * h;
                float iou = inter / (ai + mya[e] - inter + EPS_);
                if (iou > NMS_THR_)
                    atomicOr(&supp[j >> 5], 1u << (j & 31));
            }
        }
        __syncthreads();
        i = ci + 1;
    }

    // Per-thread partials: sum of kept scores, count of kept.
    float s = 0.f, c = 0.f;
    for (int e = 0; e < 8; ++e) {
        int j = e * 1024 + tid;
        unsigned w = supp[j >> 5];
        if (myk[e] >= 0.f && !((w >> (j & 31)) & 1u)) {
            s += myk[e];
            c += 1.f;
        }
    }
    psum[tid] = s;
    pcnt[tid] = c;
    __syncthreads();

    // Wave 0 (full 32-lane EXEC, as WMMA requires) reduces 1024 partials with
    // v_wmma_f32_16x16x4_f32 against an all-ones B matrix: D = A*1 + C.
    if (tid < 32) {
        v8f accS = {0.f,0.f,0.f,0.f,0.f,0.f,0.f,0.f};
        v8f accC = {0.f,0.f,0.f,0.f,0.f,0.f,0.f,0.f};
        v2f ones; ones.x = 1.f; ones.y = 1.f;
        for (int ch = 0; ch < 16; ++ch) {
            v2f a; a.x = psum[ch * 64 + tid]; a.y = psum[ch * 64 + 32 + tid];
            accS = __builtin_amdgcn_wmma_f32_16x16x4_f32(
                false, a, false, ones, (short)0, accS, false, false);
            v2f b; b.x = pcnt[ch * 64 + tid]; b.y = pcnt[ch * 64 + 32 + tid];
            accC = __builtin_amdgcn_wmma_f32_16x16x4_f32(
                false, b, false, ones, (short)0, accC, false, false);
        }
        float ts = accS[0] + accS[1] + accS[2] + accS[3] +
                   accS[4] + accS[5] + accS[6] + accS[7];
        float tc = accC[0] + accC[1] + accC[2] + accC[3] +
                   accC[4] + accC[5] + accC[6] + accC[7];
        // lanes 0-15 hold rows M=0..7 of their column, lanes 16-31 hold M=8..15
        ts += __shfl_xor(ts, 16, 32);
        tc += __shfl_xor(tc, 16, 32);
        if (tid == 0) loss_out[img] = ts / tc;
    }
}

// ---------------------------------------------------------------------------
// Host-side launcher.
//   d_in[0]: output      (B, N, 85) f32
//   d_in[1]: label_batch (B, M, 5)  f32
//   d_out  : (1, B) f32  -> 16 floats
// Workspace layout (bytes):
//   [0,        2 MB)  pred boxes   float4[B*N]
//   [2.0 MB, 2.5 MB)  keys         float [B*N]
//   [2.5 MB, 4.5 MB)  sorted boxes float4[B*N]
//   [4.5 MB, 5.0 MB)  sorted keys  float [B*N]
// ---------------------------------------------------------------------------
extern "C" void kernel_launch(void* const* d_in, const int* in_sizes, int n_in,
                              void* d_out, int out_size, void* d_ws, size_t ws_size,
                              hipStream_t stream) {
    const float* out = (const float*)d_in[0];
    const float* lab = (const float*)d_in[1];
    float* loss = (float*)d_out;

    char* ws = (char*)d_ws;
    float4* boxes  = (float4*)(ws);
    float*  key    = (float*) (ws + 2097152);   // 2.0 MB
    float4* sboxes = (float4*)(ws + 2621440);   // 2.5 MB (16B aligned)
    float*  skey   = (float*) (ws + 4718592);   // 4.5 MB

    det_prep_kernel<<<dim3((B_ * N_) / 256), dim3(256), 0, stream>>>(
        out, lab, boxes, key);
    det_sort_kernel<<<dim3(B_), dim3(1024), 0, stream>>>(
        key, boxes, skey, sboxes);
    det_nms_kernel<<<dim3(B_), dim3(1024), 0, stream>>>(
        skey, sboxes, loss);
}